// EmbeddingLoss_27582279975043
// MI455X (gfx1250) — compile-verified
//
#include <hip/hip_runtime.h>

typedef float v2f __attribute__((ext_vector_type(2)));
typedef float v8f __attribute__((ext_vector_type(8)));

#define B_    2
#define C_    32
#define N_    4096
#define TM_   256                     // 16x16 tiles per matrix dim
#define NLAB  8
#define TRI_  (TM_ * (TM_ + 1) / 2)   // 32896 upper-tri tiles per batch

// ---- zero output scalar and per-batch label counts ----
__global__ void el_init_kernel(float* out, int* counts) {
    int t = threadIdx.x;
    if (t == 0) out[0] = 0.0f;
    if (t < B_ * NLAB) counts[t] = 0;
}

// ---- label histogram: counts[b][l] ----
__global__ void el_hist_kernel(const int* __restrict__ lab, int* __restrict__ counts) {
    int idx = blockIdx.x * blockDim.x + threadIdx.x;   // [0, B*N)
    int b = idx >> 12;                                 // N = 4096
    atomicAdd(&counts[b * NLAB + lab[idx]], 1);
}

// ---- per-pixel: center over C, L2-normalize (NaN->0), store u^T [b][n][c]; w = 1/count ----
__global__ void el_norm_kernel(const float* __restrict__ emb, const int* __restrict__ lab,
                               const int* __restrict__ counts,
                               float* __restrict__ ut, float* __restrict__ w) {
    int idx = blockIdx.x * blockDim.x + threadIdx.x;   // [0, B*N)
    int b = idx >> 12;
    int n = idx & (N_ - 1);
    const float* e = emb + (size_t)b * C_ * N_ + n;    // stride-N over channels (coalesced per c)
    float x[C_];
    float mean = 0.0f;
#pragma unroll
    for (int c = 0; c < C_; ++c) { x[c] = e[(size_t)c * N_]; mean += x[c]; }
    mean *= (1.0f / C_);
    float ss = 0.0f;
#pragma unroll
    for (int c = 0; c < C_; ++c) { x[c] -= mean; ss += x[c] * x[c]; }
    float inv = (ss > 0.0f) ? (1.0f / sqrtf(ss)) : 0.0f;   // degenerate pixel -> u = 0
    float* urow = ut + (size_t)idx * C_;
#pragma unroll
    for (int c = 0; c < C_; c += 4) {
        float4 v = make_float4(x[c] * inv, x[c + 1] * inv, x[c + 2] * inv, x[c + 3] * inv);
        *(float4*)(urow + c) = v;
    }
    w[idx] = 1.0f / (float)counts[b * NLAB + lab[idx]];
}

// ---- one wave per upper-tri 16x16 tile: 8x V_WMMA_F32_16X16X4_F32 + fused epilogue ----
__global__ void __launch_bounds__(256)
el_gram_loss_kernel(const float* __restrict__ ut, const float* __restrict__ w,
                    const int* __restrict__ lab, float* __restrict__ out) {
    __shared__ float red[8];
    const int wave = blockIdx.x * 8 + (threadIdx.x >> 5);
    const int lane = threadIdx.x & 31;

    // decode wave -> (b, upper-tri tile (i,j), j >= i); all wave-uniform
    int b = wave / TRI_;
    int t = wave - b * TRI_;
    float disc = (2.0f * TM_ + 1.0f) * (2.0f * TM_ + 1.0f) - 8.0f * (float)t;
    int i = (int)(((2.0f * TM_ + 1.0f) - sqrtf(disc)) * 0.5f);
    if (i < 0) i = 0;
    if (i > TM_ - 1) i = TM_ - 1;
    // f(i) = i*TM - i*(i-1)/2 tiles precede row i; fix up float estimate
    while (i > 0 && (i * TM_ - (i * (i - 1)) / 2) > t) --i;
    while (((i + 1) * TM_ - ((i + 1) * i) / 2) <= t) ++i;
    int j = i + (t - (i * TM_ - (i * (i - 1)) / 2));

    const int hi = lane >> 4;      // K half-select for A/B fragments
    const int lo = lane & 15;      // M (A) / N (B) index
    const float* ub   = ut + (size_t)b * N_ * C_;
    const float* arow = ub + (size_t)(i * 16 + lo) * C_;  // A: rows of tile i
    const float* brow = ub + (size_t)(j * 16 + lo) * C_;  // B: rows of tile j (Gram: same layout)

    v8f acc = {};
#pragma unroll
    for (int k = 0; k < 8; ++k) {              // K = 32 in steps of 4
        int c0 = 4 * k + 2 * hi;               // VGPR0 <- K=c0, VGPR1 <- K=c0+1
        v2f a  = *(const v2f*)(arow + c0);
        v2f bb = *(const v2f*)(brow + c0);
        acc = __builtin_amdgcn_wmma_f32_16x16x4_f32(
            false, a, false, bb, (short)0, acc, false, false);
    }

    // epilogue on accumulator: C/D layout -> VGPR r, lane: row = r + 8*hi, col = lo
    const int*   lb = lab + b * N_;
    const float* wb = w   + b * N_;
    int   col = j * 16 + lo;
    int   lc  = lb[col];
    float wc  = wb[col];
    int   row0 = i * 16 + 8 * hi;
    float part = 0.0f;
#pragma unroll
    for (int r = 0; r < 8; ++r) {
        int   lr = lb[row0 + r];
        float wr = wb[row0 + r];
        float S  = 0.5f * acc[r] + 0.5f;
        float loss = (lr == lc) ? fmaxf(1.0f - S, 0.0f) : fmaxf(S - 0.5f, 0.0f);
        part += wr * wc * loss;
    }
    if (i != j) part *= 2.0f;                  // symmetry: off-diagonal tiles count twice

    // wave32 reduce, then block reduce, one atomic per block
#pragma unroll
    for (int off = 16; off > 0; off >>= 1)
        part += __shfl_down(part, off, 32);
    if (lane == 0) red[threadIdx.x >> 5] = part;
    __syncthreads();
    if (threadIdx.x == 0) {
        float s = 0.0f;
#pragma unroll
        for (int q = 0; q < 8; ++q) s += red[q];
        atomicAdd(out, s * (1.0f / (float)N_));
    }
}

extern "C" void kernel_launch(void* const* d_in, const int* in_sizes, int n_in,
                              void* d_out, int out_size, void* d_ws, size_t ws_size,
                              hipStream_t stream) {
    const float* emb = (const float*)d_in[0];   // (2, 32, 64, 64) f32
    const int*   lab = (const int*)d_in[1];     // (2, 64, 64) i32
    float* out = (float*)d_out;                 // scalar

    char* ws = (char*)d_ws;
    float* ut     = (float*)ws;                                          // B*N*C f32 (1 MB)
    float* w      = (float*)(ws + (size_t)B_ * N_ * C_ * 4);             // B*N f32
    int*   counts = (int*)  (ws + (size_t)B_ * N_ * C_ * 4
                                + (size_t)B_ * N_ * 4);                  // B*NLAB i32

    el_init_kernel<<<1, 64, 0, stream>>>(out, counts);
    el_hist_kernel<<<(B_ * N_) / 256, 256, 0, stream>>>(lab, counts);
    el_norm_kernel<<<(B_ * N_) / 256, 256, 0, stream>>>(emb, lab, counts, ut, w);

    const int nwaves = B_ * TRI_;               // 65792 waves, 8 per block
    el_gram_loss_kernel<<<nwaves / 8, 256, 0, stream>>>(ut, w, lab, out);
}